// GINConv_18141941859012
// MI455X (gfx1250) — compile-verified
//
#include <hip/hip_runtime.h>
#include <hip/hip_bf16.h>

typedef float v2f __attribute__((ext_vector_type(2)));
typedef float v8f __attribute__((ext_vector_type(8)));

#define DFEAT 64
#define EPSP1 1.5f
#define EDGES_PER_WAVE 128

// ---------------- Kernel 1: zero the aggregation workspace ----------------
__global__ void gin_zero_kernel(float4* __restrict__ p, int n4) {
    int i = blockIdx.x * blockDim.x + threadIdx.x;
    int stride = gridDim.x * blockDim.x;
    for (; i < n4; i += stride) p[i] = make_float4(0.f, 0.f, 0.f, 0.f);
}

// ---------------- Kernel 2: sorted-CSR segment scatter-add ----------------
// One wave handles EDGES_PER_WAVE contiguous edges. Lane l owns features
// [2l, 2l+1]. row_ids is sorted, so we accumulate runs in registers and flush
// with native global_atomic_add_f32 only on row change / chunk end.
__global__ void gin_scatter_kernel(const float* __restrict__ X,
                                   const int* __restrict__ col_idx,
                                   const int* __restrict__ row_ids,
                                   float* __restrict__ agg,
                                   int n_edges) {
    int gtid = blockIdx.x * blockDim.x + threadIdx.x;
    int wave = gtid >> 5;
    int lane = threadIdx.x & 31;
    long e0 = (long)wave * EDGES_PER_WAVE;
    if (e0 >= n_edges) return;
    long e1 = e0 + EDGES_PER_WAVE;
    if (e1 > n_edges) e1 = n_edges;

    const int foff = lane * 2;
    float ax = 0.f, ay = 0.f;
    int cur = row_ids[e0];
    for (long e = e0; e < e1; ++e) {
        int r = row_ids[e];   // uniform across wave -> coalesces to 1 request
        int c = col_idx[e];
        if (r != cur) {
            float* dst = agg + (size_t)cur * DFEAT + foff;
            unsafeAtomicAdd(dst,     ax);
            unsafeAtomicAdd(dst + 1, ay);
            ax = 0.f; ay = 0.f;
            cur = r;
        }
        v2f xv = *(const v2f*)(X + (size_t)c * DFEAT + foff);
        ax += xv.x;
        ay += xv.y;
    }
    float* dst = agg + (size_t)cur * DFEAT + foff;
    unsafeAtomicAdd(dst,     ax);
    unsafeAtomicAdd(dst + 1, ay);
}

// ---------------- Kernel 3: combine + GEMM via V_WMMA_F32_16X16X4_F32 -----
// One wave computes a 16x64 output tile: out[16,64] = (1.5*X + agg)[16,64] @ W[64,64].
// K=64 -> 16 WMMA k-steps of 16x16x4, for each of 4 N-tiles of 16 columns.
// W is stored in LDS in K-pair-major float2 form so each B fragment is one
// aligned ds_load_b64 (no VGPR repacking before the WMMA).
__global__ __launch_bounds__(256) void gin_wmma_gemm_kernel(
        const float* __restrict__ X,
        const float* __restrict__ agg,
        const float* __restrict__ W,
        float* __restrict__ out,
        int n_nodes) {
    // Wswz[kp*64 + col] = { W[2*kp][col], W[2*kp+1][col] }   (16 KB)
    __shared__ v2f   Wswz[(DFEAT / 2) * DFEAT];
    __shared__ float Atile[8][16 * DFEAT];       // 4 KB per wave (8 waves)

    const int tid  = threadIdx.x;
    const int wave = tid >> 5;
    const int lane = tid & 31;

    // Cooperative swizzled load of W into LDS: K-pairs become contiguous.
    for (int i = tid; i < (DFEAT / 2) * DFEAT; i += blockDim.x) {
        int kp  = i >> 6;          // K-pair index (0..31)
        int col = i & 63;
        v2f v;
        v.x = W[(2 * kp)     * DFEAT + col];
        v.y = W[(2 * kp + 1) * DFEAT + col];
        Wswz[i] = v;
    }
    __syncthreads();

    const int tile     = blockIdx.x * 8 + wave;
    const int row_base = tile * 16;
    if (row_base >= n_nodes) return;             // after the only block barrier

    // Stage combined A-tile (16 rows x 64 cols) into LDS, coalesced float4.
    float* At = Atile[wave];
    for (int i = lane; i < 16 * DFEAT / 4; i += 32) {
        int r  = i >> 4;            // 16 float4 per row
        int c4 = i & 15;
        float4 xv = ((const float4*)(X   + (size_t)(row_base + r) * DFEAT))[c4];
        float4 av = ((const float4*)(agg + (size_t)(row_base + r) * DFEAT))[c4];
        float4 v;
        v.x = EPSP1 * xv.x + av.x;
        v.y = EPSP1 * xv.y + av.y;
        v.z = EPSP1 * xv.z + av.z;
        v.w = EPSP1 * xv.w + av.w;
        ((float4*)(At + r * DFEAT))[c4] = v;
    }
    // Cross-lane dependence through LDS within this wave: LDS pipe is in-order
    // per wave; make the ordering explicit with the CDNA5 split counter.
    asm volatile("s_wait_dscnt 0" ::: "memory");

    // A fragments for all 16 k-steps (ISA 7.12.2 "32-bit A-Matrix 16x4"):
    //   lanes 0-15: M=lane, {K=kb, K=kb+1}; lanes 16-31: M=lane-16, {K=kb+2, kb+3}
    const int arow = lane & 15;
    const int koff = (lane >> 4) * 2;            // 0 or 2
    const int kph  = lane >> 4;                  // K-pair phase: 0 or 1
    v2f a[16];
#pragma unroll
    for (int ks = 0; ks < 16; ++ks) {
        int k = ks * 4 + koff;
        a[ks] = *(const v2f*)(At + arow * DFEAT + k);   // ds_load_b64
    }

    // 4 N-tiles of 16 columns each.
#pragma unroll
    for (int nt = 0; nt < 4; ++nt) {
        const int col = nt * 16 + (lane & 15);
        v8f c = {0.f, 0.f, 0.f, 0.f, 0.f, 0.f, 0.f, 0.f};
#pragma unroll
        for (int ks = 0; ks < 16; ++ks) {
            // B fragment: lane holds K-pair (k, k+1) for its column.
            // Pair index = (ks*4 + koff) / 2 = ks*2 + kph  ->  one ds_load_b64.
            v2f b = Wswz[(ks * 2 + kph) * DFEAT + col];
            c = __builtin_amdgcn_wmma_f32_16x16x4_f32(
                    /*neg_a=*/false, a[ks], /*neg_b=*/false, b,
                    /*c_mod=*/(short)0, c, /*reuse_a=*/false, /*reuse_b=*/false);
        }
        // C/D layout: VGPR v -> M = v + (lane>=16 ? 8 : 0), N = lane&15
        const int mbase = row_base + ((lane >> 4) ? 8 : 0);
#pragma unroll
        for (int v = 0; v < 8; ++v)
            out[(size_t)(mbase + v) * DFEAT + col] = c[v];
    }
}

extern "C" void kernel_launch(void* const* d_in, const int* in_sizes, int n_in,
                              void* d_out, int out_size, void* d_ws, size_t ws_size,
                              hipStream_t stream) {
    const float* X       = (const float*)d_in[0];   // [N, 64]
    const int*   col_idx = (const int*)d_in[1];     // [E]
    const int*   row_ids = (const int*)d_in[2];     // [E], sorted
    const float* W       = (const float*)d_in[3];   // [64, 64]
    float*       out     = (float*)d_out;           // [N, 64]

    const int n_nodes = in_sizes[0] / DFEAT;
    const int n_edges = in_sizes[1];

    float* agg = (float*)d_ws;                      // needs N*64*4 = 25.6 MB

    // 1) zero agg
    {
        int n4 = n_nodes * DFEAT / 4;
        int blocks = 2048;
        gin_zero_kernel<<<blocks, 256, 0, stream>>>((float4*)agg, n4);
    }
    // 2) segment scatter-add
    {
        long waves = ((long)n_edges + EDGES_PER_WAVE - 1) / EDGES_PER_WAVE;
        long threads = waves * 32;
        int blocks = (int)((threads + 255) / 256);
        gin_scatter_kernel<<<blocks, 256, 0, stream>>>(X, col_idx, row_ids, agg, n_edges);
    }
    // 3) combine + GEMM (one 16-row tile per wave, 8 waves per block)
    {
        int tiles = (n_nodes + 15) / 16;
        int blocks = (tiles + 7) / 8;
        gin_wmma_gemm_kernel<<<blocks, 256, 0, stream>>>(X, agg, W, out, n_nodes);
    }
}